// NLFEM_49838800502978
// MI455X (gfx1250) — compile-verified
//
#include <hip/hip_runtime.h>
#include <math.h>

#define NB 8
#define CH 64
#define HWP 65536
#define HW4 16384
#define EPSF 1e-5f

typedef float v2f __attribute__((ext_vector_type(2)));
typedef float v8f __attribute__((ext_vector_type(8)));

// workspace offsets (in floats)
#define OFF_K     0u          // [2][NB][HWP] key map (post-SiLU)      4 MiB
#define OFF_HALF  1048576u    // [2][NB][HWP] half activation          4 MiB
#define OFF_SUM   2097152u    // [2][NB][CH] conv sum (atomic)
#define OFF_MAX   2098176u    // [2][NB][CH] conv max (atomic)
#define OFF_FEA   2099200u    // [2][NB][CH] fea accum (atomic)
#define OFF_AVG   2100224u    // [2][NB][CH] softmax(mean)
#define OFF_MX    2101248u    // [2][NB][CH] softmax(max)
#define OFF_GATE  2102272u    // [2][NB][CH] sigmoid gate
#define OFF_KMAX  2103296u    // [2][NB]
#define OFF_KSUM  2103312u    // [2][NB]

__device__ __forceinline__ float sigmoidf_(float x){ return 1.0f/(1.0f+__expf(-x)); }
__device__ __forceinline__ float siluf_(float x){ return x*sigmoidf_(x); }

__device__ __forceinline__ void atomicMaxF(float* addr, float val){
  unsigned int* ua = (unsigned int*)addr;
  unsigned int old = *ua;
  while (__uint_as_float(old) < val){
    unsigned int assumed = old;
    old = atomicCAS(ua, assumed, __float_as_uint(val));
    if (old == assumed) break;
  }
}

__device__ __forceinline__ v8f wmma_f32x4(v2f a, v2f b, v8f c){
  // D = A(16x4 f32) x B(4x16 f32) + C(16x16 f32)
  return __builtin_amdgcn_wmma_f32_16x16x4_f32(false, a, false, b, (short)0, c, false, false);
}

__global__ void nlfem_init(float* __restrict__ ws){
  int tid = blockIdx.x*blockDim.x + threadIdx.x;
  if (tid < 1024){
    ws[OFF_SUM+tid] = 0.0f;
    ws[OFF_FEA+tid] = 0.0f;
    ws[OFF_MAX+tid] = -INFINITY;
  }
  if (tid < 16) ws[OFF_KMAX+tid] = -INFINITY;
}

// Pass 1: conv1x1(conv_w) stats (sum/max per channel) + key map + kmax.
__global__ __launch_bounds__(256) void nlfem_pass1(
    const float* __restrict__ rgb, const float* __restrict__ ir,
    const float* __restrict__ conv_w,
    const float* __restrict__ key_w, const float* __restrict__ key_g,
    const float* __restrict__ key_b, const float* __restrict__ key_m,
    const float* __restrict__ key_v,
    float* __restrict__ ws)
{
  __shared__ __align__(16) float afrag[4096]; // [4 mt][16 ks][32 lane][2]
  __shared__ __align__(16) float keyfrag[64]; // [16 ks][2 hf][2]
  const int tid = threadIdx.x;
  const int n = blockIdx.y, side = blockIdx.z;
  const float sk  = key_g[0]*rsqrtf(key_v[0]+EPSF);
  const float shk = key_b[0]-key_m[0]*sk;

  for (int idx = tid; idx < 4096; idx += 256){
    int mt = idx>>10, rem = idx&1023, ks = rem>>6, l = (rem>>1)&31, comp = idx&1;
    int row = (mt<<4)+(l&15); int kk = (ks<<2)+((l>>4)<<1)+comp;
    afrag[idx] = conv_w[row*64+kk];
  }
  if (tid < 64){
    int ks = tid>>2, h = (tid>>1)&1, comp = tid&1;
    keyfrag[tid] = key_w[(ks<<2)+(h<<1)+comp]*sk;
  }
  __syncthreads();

  const int lane = tid & 31, hf = lane>>4, pix = lane&15;
  const int wid = blockIdx.x*8 + (tid>>5);
  const int nw  = gridDim.x*8;
  const float* x  = side ? ir : rgb;
  const float* xn = x + (size_t)n*CH*HWP;
  float* kst = ws + OFF_K + (size_t)(side*NB+n)*HWP;

  float ssum[32], smax[32];
  #pragma unroll
  for (int i=0;i<32;++i){ ssum[i]=0.0f; smax[i]=-INFINITY; }
  float kmr = -INFINITY;

  for (int t = wid; t < HWP/16; t += nw){
    const int p0 = t<<4;
    const float* xb = xn + p0 + pix;
    v8f acc[4];
    #pragma unroll
    for (int mt=0;mt<4;++mt)
      #pragma unroll
      for (int j=0;j<8;++j) acc[mt][j] = 0.0f;
    float kdot = 0.0f;
    #pragma unroll
    for (int ks=0; ks<16; ++ks){
      const int c0 = (ks<<2)+(hf<<1);
      v2f b; b.x = xb[(size_t)c0*HWP]; b.y = xb[(size_t)(c0+1)*HWP];
      v2f kw = *(const v2f*)(keyfrag + (ks<<2)+(hf<<1));
      kdot += kw.x*b.x + kw.y*b.y;
      #pragma unroll
      for (int mt=0; mt<4; ++mt){
        v2f a = *(const v2f*)(afrag + (((mt<<4)+ks)<<6) + (lane<<1));
        acc[mt] = wmma_f32x4(a, b, acc[mt]);
      }
    }
    #pragma unroll
    for (int mt=0;mt<4;++mt)
      #pragma unroll
      for (int r=0;r<8;++r){
        float v = acc[mt][r];
        ssum[mt*8+r] += v;
        smax[mt*8+r]  = fmaxf(smax[mt*8+r], v);
      }
    float kd2 = kdot + __shfl_xor(kdot, 16, 32);
    float kv  = siluf_(kd2 + shk);
    if (lane < 16) kst[p0 + pix] = kv;
    kmr = fmaxf(kmr, kv);
  }

  const int base = (side*NB+n)*CH;
  #pragma unroll
  for (int i=0;i<32;++i){
    float s = ssum[i], m = smax[i];
    s += __shfl_xor(s,1,32); s += __shfl_xor(s,2,32);
    s += __shfl_xor(s,4,32); s += __shfl_xor(s,8,32);
    m = fmaxf(m,__shfl_xor(m,1,32)); m = fmaxf(m,__shfl_xor(m,2,32));
    m = fmaxf(m,__shfl_xor(m,4,32)); m = fmaxf(m,__shfl_xor(m,8,32));
    if ((lane&15)==0){
      int M = ((i>>3)<<4) + (i&7) + (hf<<3);
      atomicAdd(ws + OFF_SUM + base + M, s);
      atomicMaxF(ws + OFF_MAX + base + M, m);
    }
  }
  kmr = fmaxf(kmr,__shfl_xor(kmr,1,32));
  kmr = fmaxf(kmr,__shfl_xor(kmr,2,32));
  kmr = fmaxf(kmr,__shfl_xor(kmr,4,32));
  kmr = fmaxf(kmr,__shfl_xor(kmr,8,32));
  kmr = fmaxf(kmr,__shfl_xor(kmr,16,32));
  if (lane==0) atomicMaxF(ws + OFF_KMAX + side*NB + n, kmr);
}

// Pass 2: finalize softmaxes + key softmax denominator.
__global__ __launch_bounds__(256) void nlfem_pass2(const float* __restrict__ conv_b,
                                                   float* __restrict__ ws){
  const int bid = blockIdx.x; const int side = bid>>3, n = bid&7;
  const int base = (side*NB+n)*CH;
  __shared__ float a1[64], a2[64], red[256], st[4];
  const int tid = threadIdx.x;
  if (tid < 64){
    a1[tid] = ws[OFF_SUM+base+tid]*(1.0f/65536.0f) + conv_b[tid];
    a2[tid] = ws[OFF_MAX+base+tid] + conv_b[tid];
  }
  __syncthreads();
  if (tid == 0){
    float m1=-INFINITY,m2=-INFINITY;
    for (int i=0;i<64;++i){ m1=fmaxf(m1,a1[i]); m2=fmaxf(m2,a2[i]); }
    float s1=0.0f,s2=0.0f;
    for (int i=0;i<64;++i){ s1+=__expf(a1[i]-m1); s2+=__expf(a2[i]-m2); }
    st[0]=m1; st[1]=s1; st[2]=m2; st[3]=s2;
  }
  __syncthreads();
  if (tid < 64){
    ws[OFF_AVG+base+tid] = __expf(a1[tid]-st[0])/st[1];
    ws[OFF_MX +base+tid] = __expf(a2[tid]-st[2])/st[3];
  }
  const float kmaxv = ws[OFF_KMAX + side*NB + n];
  const float* kp = ws + OFF_K + (size_t)(side*NB+n)*HWP;
  float loc = 0.0f;
  for (int i=tid;i<HWP;i+=256) loc += __expf(kp[i]-kmaxv);
  red[tid]=loc; __syncthreads();
  for (int s2=128; s2>0; s2>>=1){ if (tid<s2) red[tid]+=red[tid+s2]; __syncthreads(); }
  if (tid==0) ws[OFF_KSUM + side*NB + n] = red[0];
}

// Pass 3: recompute value conv of side s (WMMA), emit fea partials and half
// activation for the opposite output side o = 1-s.
__global__ __launch_bounds__(256) void nlfem_pass3(
    const float* __restrict__ rgb, const float* __restrict__ ir,
    const float* __restrict__ val_w, const float* __restrict__ val_g,
    const float* __restrict__ val_b, const float* __restrict__ val_m,
    const float* __restrict__ val_v,
    const float* __restrict__ half_w, const float* __restrict__ half_g,
    const float* __restrict__ half_b, const float* __restrict__ half_m,
    const float* __restrict__ half_v,
    float* __restrict__ ws)
{
  __shared__ __align__(16) float afrag[4096];
  __shared__ __align__(16) float pch[256]; // [M]{shv, avg, mx, 0}
  const int tid = threadIdx.x;
  const int n = blockIdx.y, s = blockIdx.z, o = 1 - s;

  for (int idx = tid; idx < 4096; idx += 256){
    int mt = idx>>10, rem = idx&1023, ks = rem>>6, l = (rem>>1)&31, comp = idx&1;
    int row = (mt<<4)+(l&15); int kk = (ks<<2)+((l>>4)<<1)+comp;
    float sv = val_g[row]*rsqrtf(val_v[row]+EPSF);
    afrag[idx] = val_w[row*64+kk]*sv;  // fold BN scale into weights
  }
  if (tid < 64){
    float sv = val_g[tid]*rsqrtf(val_v[tid]+EPSF);
    pch[tid*4+0] = val_b[tid]-val_m[tid]*sv;
    pch[tid*4+1] = ws[OFF_AVG + (o*NB+n)*CH + tid];
    pch[tid*4+2] = ws[OFF_MX  + (o*NB+n)*CH + tid];
    pch[tid*4+3] = 0.0f;
  }
  __syncthreads();

  const int lane = tid&31, hf = lane>>4, pix = lane&15;
  const int wid = blockIdx.x*8 + (tid>>5);
  const int nw  = gridDim.x*8;
  const float* x  = s ? ir : rgb;
  const float* xn = x + (size_t)n*CH*HWP;
  const float* kpo = ws + OFF_K + (size_t)(o*NB+n)*HWP;
  float* hst = ws + OFF_HALF + (size_t)(o*NB+n)*HWP;
  const float kmaxo = ws[OFF_KMAX + o*NB + n];
  const float sh  = half_g[0]*rsqrtf(half_v[0]+EPSF);
  const float h0  = half_w[0]*sh, h1 = half_w[1]*sh;
  const float shh = half_b[0]-half_m[0]*sh;

  float facc[32];
  #pragma unroll
  for (int i=0;i<32;++i) facc[i]=0.0f;

  for (int t = wid; t < HWP/16; t += nw){
    const int p0 = t<<4;
    const float* xb = xn + p0 + pix;
    v8f acc[4];
    #pragma unroll
    for (int mt=0;mt<4;++mt)
      #pragma unroll
      for (int j=0;j<8;++j) acc[mt][j] = 0.0f;
    #pragma unroll
    for (int ks=0; ks<16; ++ks){
      const int c0 = (ks<<2)+(hf<<1);
      v2f b; b.x = xb[(size_t)c0*HWP]; b.y = xb[(size_t)(c0+1)*HWP];
      #pragma unroll
      for (int mt=0; mt<4; ++mt){
        v2f a = *(const v2f*)(afrag + (((mt<<4)+ks)<<6) + (lane<<1));
        acc[mt] = wmma_f32x4(a, b, acc[mt]);
      }
    }
    float wk = __expf(kpo[p0+pix] - kmaxo);  // un-normalized key weight
    float am = 0.0f, mm = 0.0f;
    #pragma unroll
    for (int mt=0;mt<4;++mt)
      #pragma unroll
      for (int r=0;r<8;++r){
        int M = (mt<<4)+r+(hf<<3);
        float4 p = *(const float4*)(pch + M*4);
        float vv = siluf_(acc[mt][r] + p.x);
        facc[mt*8+r] += vv*wk;
        am += p.y*vv; mm += p.z*vv;
      }
    am += __shfl_xor(am,16,32);
    mm += __shfl_xor(mm,16,32);
    float hv = siluf_(h0*am + h1*mm + shh);
    if (lane < 16) hst[p0+pix] = hv;
  }
  const int base = (o*NB+n)*CH;
  #pragma unroll
  for (int i=0;i<32;++i){
    float f = facc[i];
    f += __shfl_xor(f,1,32); f += __shfl_xor(f,2,32);
    f += __shfl_xor(f,4,32); f += __shfl_xor(f,8,32);
    if ((lane&15)==0){
      int M = ((i>>3)<<4)+(i&7)+(hf<<3);
      atomicAdd(ws + OFF_FEA + base + M, f);
    }
  }
}

// Pass 4: gate = sigmoid(LN(fea @ convb_w^T))
__global__ void nlfem_pass4(const float* __restrict__ convb_w,
                            const float* __restrict__ ln_g, const float* __restrict__ ln_b,
                            float* __restrict__ ws){
  const int bid = blockIdx.x; const int side = bid>>3, n = bid&7;
  const int base = (side*NB+n)*CH;
  __shared__ float fsh[64], zsh[64], st[2];
  const int tid = threadIdx.x; // 64 threads
  float feaf = ws[OFF_FEA+base+tid] / ws[OFF_KSUM + side*NB + n];
  fsh[tid] = feaf; __syncthreads();
  float z = 0.0f;
  for (int c=0;c<64;++c) z += fsh[c]*convb_w[tid*64+c];
  zsh[tid] = z; __syncthreads();
  if (tid == 0){
    float mu=0.0f; for (int i=0;i<64;++i) mu+=zsh[i]; mu *= (1.0f/64.0f);
    float var=0.0f; for (int i=0;i<64;++i){ float d=zsh[i]-mu; var+=d*d; } var *= (1.0f/64.0f);
    st[0]=mu; st[1]=rsqrtf(var+EPSF);
  }
  __syncthreads();
  float g = (z-st[0])*st[1]*ln_g[tid]+ln_b[tid];
  ws[OFF_GATE+base+tid] = sigmoidf_(g);
}

// Pass 5: out = gate * half + x (float4 streaming)
__global__ __launch_bounds__(256) void nlfem_pass5(
  const float* __restrict__ rgb, const float* __restrict__ ir,
  const float* __restrict__ ws, float* __restrict__ out){
  const size_t TOT4 = (size_t)2*NB*CH*HW4;
  const float4* r4 = (const float4*)rgb;
  const float4* i4 = (const float4*)ir;
  const float4* h4 = (const float4*)(ws + OFF_HALF);
  float4* o4 = (float4*)out;
  for (size_t i = (size_t)blockIdx.x*blockDim.x + threadIdx.x; i < TOT4;
       i += (size_t)gridDim.x*blockDim.x){
    const int t = (int)((i>>23)&1);
    const int n = (int)((i>>20)&7);
    const int c = (int)((i>>14)&63);
    const size_t p4 = i & 16383u;
    float4 xv = (t ? i4 : r4)[i & ((1u<<23)-1u)];
    float4 hv = h4[((size_t)(t*NB+n)<<14) + p4];
    float g = ws[OFF_GATE + ((t*NB+n)<<6) + c];
    float4 ov;
    ov.x = g*hv.x + xv.x; ov.y = g*hv.y + xv.y;
    ov.z = g*hv.z + xv.z; ov.w = g*hv.w + xv.w;
    o4[i] = ov;
  }
}

extern "C" void kernel_launch(void* const* d_in, const int* in_sizes, int n_in,
                              void* d_out, int out_size, void* d_ws, size_t ws_size,
                              hipStream_t stream){
  const float* rgb    = (const float*)d_in[0];
  const float* ir     = (const float*)d_in[1];
  const float* conv_w = (const float*)d_in[2];
  const float* conv_b = (const float*)d_in[3];
  const float* key_w  = (const float*)d_in[4];
  const float* key_g  = (const float*)d_in[5];
  const float* key_b  = (const float*)d_in[6];
  const float* key_m  = (const float*)d_in[7];
  const float* key_v  = (const float*)d_in[8];
  const float* val_w  = (const float*)d_in[9];
  const float* val_g  = (const float*)d_in[10];
  const float* val_b  = (const float*)d_in[11];
  const float* val_m  = (const float*)d_in[12];
  const float* val_v  = (const float*)d_in[13];
  const float* convb_w= (const float*)d_in[14];
  const float* half_w = (const float*)d_in[15];
  const float* half_g = (const float*)d_in[16];
  const float* half_b = (const float*)d_in[17];
  const float* half_m = (const float*)d_in[18];
  const float* half_v = (const float*)d_in[19];
  const float* ln_g   = (const float*)d_in[20];
  const float* ln_b   = (const float*)d_in[21];
  float* ws  = (float*)d_ws;
  float* out = (float*)d_out;

  nlfem_init<<<4,256,0,stream>>>(ws);
  nlfem_pass1<<<dim3(64,NB,2),256,0,stream>>>(rgb, ir, conv_w,
                                              key_w, key_g, key_b, key_m, key_v, ws);
  nlfem_pass2<<<16,256,0,stream>>>(conv_b, ws);
  nlfem_pass3<<<dim3(64,NB,2),256,0,stream>>>(rgb, ir, val_w, val_g, val_b, val_m, val_v,
                                              half_w, half_g, half_b, half_m, half_v, ws);
  nlfem_pass4<<<16,64,0,stream>>>(convb_w, ln_g, ln_b, ws);
  nlfem_pass5<<<4096,256,0,stream>>>(rgb, ir, ws, out);
}